// MRConv2d_81638738362644
// MI455X (gfx1250) — compile-verified
//
#include <hip/hip_runtime.h>

typedef float v2f __attribute__((ext_vector_type(2)));
typedef float v8f __attribute__((ext_vector_type(8)));

#define NNODES   25000
#define CIN      128      // input channels of x
#define OUTCH    256
#define KNB      16       // neighbors
#define TILE_N   16
#define EPS      1e-5f

// ---------------------------------------------------------------------------
// Kernel 1: fused gather + max-relative + grouped 1x1 conv (WMMA f32) + bias.
// Block: 128 threads (4 waves), one tile of 16 nodes. Wave w = group g:
//   y_g(64x16) = W_g(64x64) * h_g(64x16)
// kb-outer / mt-inner with 4 live accumulators: each LDS B-fragment feeds
// 4 consecutive v_wmma_f32_16x16x4_f32 ops (64 WMMAs per wave total).
// No cross-lane reductions here: BN stats are a separate L2-resident pass.
// ---------------------------------------------------------------------------
__global__ __launch_bounds__(128) void fused_mrconv_kernel(
    const float* __restrict__ x,      // (128, 25000)
    const float* __restrict__ Wt,     // (4, 64, 64)  [g][o][i]
    const float* __restrict__ bias,   // (256)
    const int*   __restrict__ eidx,   // (2, 25000, 16)
    float* __restrict__ y)            // (256, 25000)
{
    __shared__ float hs[256 * TILE_N];      // h tile [ch][n], ch interleaved (x, rel)
    __shared__ int   sidx0[TILE_N * KNB];
    __shared__ int   sidx1[TILE_N * KNB];

    const int tid   = threadIdx.x;
    const int node0 = blockIdx.x * TILE_N;

    const int g    = tid >> 5;          // wave id = group id
    const int lane = tid & 31;
    const int lh   = lane >> 4;         // lane half (0/1)
    const int ln   = lane & 15;         // column n (B/C/D) or row m (A)
    const float* Wg = Wt + g * 64 * 64;

    // warm the cache for this wave's weight slice (global_prefetch_b8)
    __builtin_prefetch(Wg + lane * 128, 0, 1);

    // ---- stage indices for the 16 nodes into LDS ----
    for (int i = tid; i < TILE_N * KNB; i += 128) {
        int n  = i >> 4;             // node within tile
        int k  = i & 15;
        int gn = node0 + n;
        int gc = gn < NNODES ? gn : (NNODES - 1);
        sidx0[i] = eidx[gc * KNB + k];                     // edge_index[0] -> x_j
        sidx1[i] = eidx[NNODES * KNB + gc * KNB + k];      // edge_index[1] -> x_i
    }
    __syncthreads();

    // ---- build h tile: h[2c] = x[c,n], h[2c+1] = max_k(x[c,j]-x[c,i]) ----
    for (int p = tid; p < CIN * TILE_N; p += 128) {
        int c  = p >> 4;
        int n  = p & 15;
        int gn = node0 + n;
        bool valid = gn < NNODES;
        const float* xc = x + (size_t)c * NNODES;
        float xv = valid ? xc[gn] : 0.0f;
        float m = -__builtin_inff();
        #pragma unroll
        for (int k = 0; k < KNB; ++k) {
            float d = xc[sidx0[n * KNB + k]] - xc[sidx1[n * KNB + k]];
            m = fmaxf(m, d);
        }
        hs[(2 * c)     * TILE_N + n] = xv;
        hs[(2 * c + 1) * TILE_N + n] = valid ? m : 0.0f;
    }
    __syncthreads();

    // ---- WMMA grouped GEMM ----
    // init accumulators with bias: C[r] holds (M = mt*16 + r + 8*lh, N = ln)
    v8f acc0, acc1, acc2, acc3;
    #pragma unroll
    for (int r = 0; r < 8; ++r) {
        acc0[r] = bias[g * 64 +  0 + r + 8 * lh];
        acc1[r] = bias[g * 64 + 16 + r + 8 * lh];
        acc2[r] = bias[g * 64 + 32 + r + 8 * lh];
        acc3[r] = bias[g * 64 + 48 + r + 8 * lh];
    }

    #pragma unroll
    for (int kb = 0; kb < 16; ++kb) {
        const int k0 = kb * 4 + 2 * lh;   // this lane's K pair base
        // B 4x16 f32: lane holds N = ln, K = 2*lh + v  (shared by all 4 m-tiles)
        v2f bf;
        bf[0] = hs[(g * 64 + k0)     * TILE_N + ln];
        bf[1] = hs[(g * 64 + k0 + 1) * TILE_N + ln];
        // A 16x4 f32: lane holds M = ln, K = 2*lh + v
        v2f a0, a1, a2, a3;
        a0[0] = Wg[( 0 + ln) * 64 + k0];  a0[1] = Wg[( 0 + ln) * 64 + k0 + 1];
        a1[0] = Wg[(16 + ln) * 64 + k0];  a1[1] = Wg[(16 + ln) * 64 + k0 + 1];
        a2[0] = Wg[(32 + ln) * 64 + k0];  a2[1] = Wg[(32 + ln) * 64 + k0 + 1];
        a3[0] = Wg[(48 + ln) * 64 + k0];  a3[1] = Wg[(48 + ln) * 64 + k0 + 1];
        acc0 = __builtin_amdgcn_wmma_f32_16x16x4_f32(false, a0, false, bf, (short)0, acc0, false, false);
        acc1 = __builtin_amdgcn_wmma_f32_16x16x4_f32(false, a1, false, bf, (short)0, acc1, false, false);
        acc2 = __builtin_amdgcn_wmma_f32_16x16x4_f32(false, a2, false, bf, (short)0, acc2, false, false);
        acc3 = __builtin_amdgcn_wmma_f32_16x16x4_f32(false, a3, false, bf, (short)0, acc3, false, false);
    }

    // ---- store y (lanes 0-15: row m, cols node0..+15 -> 64B segments) ----
    const int gn_col = node0 + ln;
    if (gn_col < NNODES) {
        const int chb = g * 64 + 8 * lh;
        #pragma unroll
        for (int r = 0; r < 8; ++r) {
            y[(size_t)(chb +  0 + r) * NNODES + gn_col] = acc0[r];
            y[(size_t)(chb + 16 + r) * NNODES + gn_col] = acc1[r];
            y[(size_t)(chb + 32 + r) * NNODES + gn_col] = acc2[r];
            y[(size_t)(chb + 48 + r) * NNODES + gn_col] = acc3[r];
        }
    }
}

// ---------------------------------------------------------------------------
// Kernel 2: per-channel BN stats over y (one block per channel, coalesced
// row scan, LDS tree reduce) -> scale/shift. y is L2-resident (25.6MB/192MB).
// ---------------------------------------------------------------------------
__global__ __launch_bounds__(256) void bn_reduce_kernel(
    const float* __restrict__ y,
    const float* __restrict__ gamma,
    const float* __restrict__ beta,
    float* __restrict__ scale,
    float* __restrict__ shift)
{
    __shared__ float ssum[256];
    __shared__ float ssqs[256];
    const int c = blockIdx.x;
    const float* yr = y + (size_t)c * NNODES;
    float s = 0.0f, q = 0.0f;
    for (int n = threadIdx.x; n < NNODES; n += 256) {
        float v = yr[n];
        s += v;
        q = fmaf(v, v, q);
    }
    ssum[threadIdx.x] = s;
    ssqs[threadIdx.x] = q;
    __syncthreads();
    #pragma unroll
    for (int off = 128; off > 0; off >>= 1) {
        if (threadIdx.x < off) {
            ssum[threadIdx.x] += ssum[threadIdx.x + off];
            ssqs[threadIdx.x] += ssqs[threadIdx.x + off];
        }
        __syncthreads();
    }
    if (threadIdx.x == 0) {
        const float inv = 1.0f / (float)NNODES;
        float mean = ssum[0] * inv;
        float var  = ssqs[0] * inv - mean * mean;
        float sc   = gamma[c] * rsqrtf(var + EPS);
        scale[c] = sc;
        shift[c] = beta[c] - mean * sc;
    }
}

// ---------------------------------------------------------------------------
// Kernel 3: out = relu(y*scale + shift), coalesced elementwise
// ---------------------------------------------------------------------------
__global__ __launch_bounds__(256) void bn_apply_kernel(
    const float* __restrict__ y,
    const float* __restrict__ scale,
    const float* __restrict__ shift,
    float* __restrict__ out)
{
    int i = blockIdx.x * 256 + threadIdx.x;
    if (i < OUTCH * NNODES) {
        int c = i / NNODES;
        float v = fmaf(y[i], scale[c], shift[c]);
        out[i] = v > 0.0f ? v : 0.0f;
    }
}

// ---------------------------------------------------------------------------
extern "C" void kernel_launch(void* const* d_in, const int* in_sizes, int n_in,
                              void* d_out, int out_size, void* d_ws, size_t ws_size,
                              hipStream_t stream) {
    const float* x     = (const float*)d_in[0];   // (1,128,25000,1)
    const float* Wt    = (const float*)d_in[1];   // (4,64,64)
    const float* bias  = (const float*)d_in[2];   // (256)
    const float* gamma = (const float*)d_in[3];   // (256)
    const float* beta  = (const float*)d_in[4];   // (256)
    const int*   eidx  = (const int*)d_in[5];     // (2,1,25000,16)

    float* y     = (float*)d_ws;                  // 256*25000 floats
    float* scale = y + (size_t)OUTCH * NNODES;    // 256
    float* shift = scale + OUTCH;                 // 256
    float* out   = (float*)d_out;

    int nblocks = (NNODES + TILE_N - 1) / TILE_N; // 1563
    fused_mrconv_kernel<<<nblocks, 128, 0, stream>>>(x, Wt, bias, eidx, y);

    bn_reduce_kernel<<<OUTCH, 256, 0, stream>>>(y, gamma, beta, scale, shift);

    int total = OUTCH * NNODES;
    bn_apply_kernel<<<(total + 255) / 256, 256, 0, stream>>>(y, scale, shift, out);
}